// Model_88914412961847
// MI455X (gfx1250) — compile-verified
//
#include <hip/hip_runtime.h>
#include <cstdint>

#define NN   100000   // nodes (16 * 6250 exactly -> GEMM grid is exact)
#define FIN  256      // input features
#define HD   128      // hidden dim
#define KH   10       // propagation hops for h1

typedef __attribute__((ext_vector_type(2))) float v2f;
typedef __attribute__((ext_vector_type(8))) float v8f;

// relaxed agent-scope fp32 atomic add -> single GLOBAL_ATOMIC_ADD_F32 (no return)
__device__ __forceinline__ void fadd_relaxed(float* p, float v) {
  __hip_atomic_fetch_add(p, v, __ATOMIC_RELAXED, __HIP_MEMORY_SCOPE_AGENT);
}

// ---------------- GCN norm: deg -> rsqrt ----------------
__global__ void k_deg_init(float* deg, int n) {
  int i = blockIdx.x * blockDim.x + threadIdx.x;
  if (i < n) deg[i] = 1.0f;                       // fresh self-loop weight
}

__global__ void k_deg_edges(const int* __restrict__ s, const int* __restrict__ r,
                            float* deg, int e) {
  int i = blockIdx.x * blockDim.x + threadIdx.x;
  if (i < e) {
    int ss = s[i], rr = r[i];
    if (ss != rr) fadd_relaxed(&deg[rr], 1.0f);   // existing self-loops get weight 0
  }
}

__global__ void k_deg_rsqrt(float* deg, int n) {
  int i = blockIdx.x * blockDim.x + threadIdx.x;
  if (i < n) {
    float d = deg[i];
    deg[i] = d > 0.0f ? rsqrtf(d) : 0.0f;
  }
}

// ---------------- fp32 WMMA GEMM: Y[N,128] = X[N,256] @ W[256,128] + b ----------------
// 8 waves per block; wave w owns output tile rows [16*bx,16*bx+16) x cols [16w,16w+16).
// K-loop of 64 x v_wmma_f32_16x16x4_f32 (exact fp32 on the matrix pipe).
// Grid is exact (N % 16 == 0), so no row guards: EXEC stays all-1s (WMMA requirement)
// and the epilogue is 8 unconditional b32 stores.
__global__ __launch_bounds__(256) void k_gemm_bias(
    const float* __restrict__ X, const float* __restrict__ W,
    const float* __restrict__ b, float* __restrict__ Y)
{
  const int lane  = threadIdx.x & 31;
  const int wave  = threadIdx.x >> 5;
  const int lm    = lane & 15;        // M (A) / N (B) index within tile
  const int hi    = lane >> 4;        // half-wave selects K pair
  const int khalf = hi * 2;
  const int row0  = blockIdx.x * 16;
  const int col   = wave * 16 + lm;

  const float* xrow = X + (size_t)(row0 + lm) * FIN;

  v8f acc = {};
  for (int k = 0; k < FIN; k += 4) {
    // A 16x4: lanes0-15 hold K=k+0,k+1 ; lanes16-31 hold K=k+2,k+3
    const float2 av = *(const float2*)(xrow + k + khalf);
    v2f a; a.x = av.x; a.y = av.y;
    // B 4x16: mirrored layout, N across lanes, K pair per half-wave
    v2f bm;
    bm.x = W[(size_t)(k + khalf)     * HD + col];
    bm.y = W[(size_t)(k + khalf + 1) * HD + col];
    acc = __builtin_amdgcn_wmma_f32_16x16x4_f32(
        /*neg_a=*/false, a, /*neg_b=*/false, bm,
        /*c_mod=*/(short)0, acc, /*reuse_a=*/false, /*reuse_b=*/false);
  }

  const float bias = b[col];
  float* yout = Y + (size_t)(row0 + hi * 8) * HD + col;
#pragma unroll
  for (int v = 0; v < 8; ++v)         // D: VGPR v = row (v + hi*8), N = lane&15
    yout[(size_t)v * HD] = acc[v] + bias;
}

// ---------------- propagation: out = D^-1/2 A D^-1/2 h (with fresh self-loops) ---------
// pass 1: self-loop term, plain stores (also initializes out)
__global__ __launch_bounds__(256) void k_prop_self(
    const float* __restrict__ h, const float* __restrict__ dinv,
    float* __restrict__ out, int n)
{
  long long idx = (long long)blockIdx.x * blockDim.x + threadIdx.x; // n*32 lanes
  if (idx >= (long long)n * 32) return;
  int node = (int)(idx >> 5);
  int c    = ((int)idx & 31) * 4;
  float d  = dinv[node];
  float s  = d * d;
  float4 v = *(const float4*)(h + (size_t)node * HD + c);
  float4 o; o.x = v.x * s; o.y = v.y * s; o.z = v.z * s; o.w = v.w * s;
  *(float4*)(out + (size_t)node * HD + c) = o;
}

// pass 2: one wave per edge, float4 gather + 4x GLOBAL_ATOMIC_ADD_F32 scatter
__global__ __launch_bounds__(256) void k_prop_edges(
    const int* __restrict__ es, const int* __restrict__ er,
    const float* __restrict__ dinv, const float* __restrict__ h,
    float* __restrict__ out, int e)
{
  const int lane = threadIdx.x & 31;
  int ei = blockIdx.x * 8 + (threadIdx.x >> 5);
  if (ei >= e) return;
  int s = es[ei], r = er[ei];
  if (s == r) return;                               // self-edges have weight 0
  float nrm = dinv[s] * dinv[r];
  float4 v = *(const float4*)(h + (size_t)s * HD + lane * 4);
  float* o = out + (size_t)r * HD + lane * 4;
  fadd_relaxed(o + 0, v.x * nrm);
  fadd_relaxed(o + 1, v.y * nrm);
  fadd_relaxed(o + 2, v.z * nrm);
  fadd_relaxed(o + 3, v.w * nrm);
}

// ---------------- h1 epilogue: score = sigmoid(pps . W_p + b_p) ----------------
// one wave per (node, hop): float4 partial dot + shuffle reduce
__global__ __launch_bounds__(256) void k_score(
    const float* __restrict__ pps, const float* __restrict__ Wp,
    const float* __restrict__ bp, float* __restrict__ score, int n)
{
  const int lane = threadIdx.x & 31;
  int w = blockIdx.x * 8 + (threadIdx.x >> 5);
  if (w >= n * (KH + 1)) return;
  int node = w / (KH + 1);
  int k    = w % (KH + 1);
  float4 v  = *(const float4*)(pps + ((size_t)k * NN + node) * HD + lane * 4);
  float4 wp = *(const float4*)(Wp + lane * 4);
  float dot = v.x * wp.x + v.y * wp.y + v.z * wp.z + v.w * wp.w;
#pragma unroll
  for (int off = 16; off > 0; off >>= 1) dot += __shfl_down(dot, off, 32);
  if (lane == 0) {
    float t = dot + bp[0];
    score[(size_t)node * (KH + 1) + k] = 1.0f / (1.0f + expf(-t));
  }
}

// out_h1[n,c] = sum_k score[n,k] * pps[k,n,c]
__global__ __launch_bounds__(128) void k_combine(
    const float* __restrict__ pps, const float* __restrict__ score,
    float* __restrict__ out, int n)
{
  int node = blockIdx.x;
  int c    = threadIdx.x;
  __shared__ float sc[KH + 1];
  if (c < KH + 1) sc[c] = score[(size_t)node * (KH + 1) + c];
  __syncthreads();
  float acc = 0.0f;
#pragma unroll
  for (int k = 0; k <= KH; ++k)
    acc += sc[k] * pps[((size_t)k * NN + node) * HD + c];
  out[(size_t)node * HD + c] = acc;
}

// ---------------- z1 = z0*w + zk*(1-w), w = z0 . W_lin + b_lin ----------------
__global__ __launch_bounds__(256) void k_z1(
    const float* __restrict__ z0, const float* __restrict__ zk,
    const float* __restrict__ Wl, const float* __restrict__ bl,
    float* __restrict__ out, int n)
{
  const int lane = threadIdx.x & 31;
  int node = blockIdx.x * 8 + (threadIdx.x >> 5);
  if (node >= n) return;
  float4 z  = *(const float4*)(z0 + (size_t)node * HD + lane * 4);
  float4 wl = *(const float4*)(Wl + lane * 4);
  float dot = z.x * wl.x + z.y * wl.y + z.z * wl.z + z.w * wl.w;
#pragma unroll
  for (int off = 16; off > 0; off >>= 1) dot += __shfl_xor(dot, off, 32);
  float w = dot + bl[0];
  float4 kv = *(const float4*)(zk + (size_t)node * HD + lane * 4);
  float4 o;
  o.x = z.x * w + kv.x * (1.0f - w);
  o.y = z.y * w + kv.y * (1.0f - w);
  o.z = z.z * w + kv.z * (1.0f - w);
  o.w = z.w * w + kv.w * (1.0f - w);
  *(float4*)(out + (size_t)node * HD + lane * 4) = o;
}

static inline int cdiv_i(long long a, long long b) { return (int)((a + b - 1) / b); }

extern "C" void kernel_launch(void* const* d_in, const int* in_sizes, int n_in,
                              void* d_out, int out_size, void* d_ws, size_t ws_size,
                              hipStream_t stream) {
  const float* x     = (const float*)d_in[0];
  const int*   ei    = (const int*)d_in[1];
  const int*   knn   = (const int*)d_in[2];
  const float* W_s1  = (const float*)d_in[3];
  const float* b_s1  = (const float*)d_in[4];
  const float* W_f2  = (const float*)d_in[5];
  const float* b_f2  = (const float*)d_in[6];
  const float* W_l1  = (const float*)d_in[7];
  const float* b_l1  = (const float*)d_in[8];
  const float* W_p   = (const float*)d_in[9];
  const float* b_p   = (const float*)d_in[10];
  const float* W_lin = (const float*)d_in[11];
  const float* b_lin = (const float*)d_in[12];

  const int N  = NN;
  const int Ee = in_sizes[1] / 2;
  const int Ek = in_sizes[2] / 2;
  const int* e_s = ei;   const int* e_r = ei  + Ee;
  const int* k_s = knn;  const int* k_r = knn + Ek;

  const size_t NHf = (size_t)N * HD;
  float* ws     = (float*)d_ws;
  float* dinv_e = ws;                                // [N]
  float* dinv_k = ws + N;                            // [N]
  float* ping   = ws + 2 * (size_t)N;                // [N,128]
  float* pong   = ping + NHf;                        // [N,128]
  float* score  = pong + NHf;                        // [N,11]
  float* pps    = score + (size_t)N * (KH + 1);      // [11,N,128]

  float* out_h0 = (float*)d_out;
  float* out_h1 = out_h0 + NHf;
  float* out_z0 = out_h1 + NHf;
  float* out_z1 = out_z0 + NHf;

  const int nodeBlks = cdiv_i(N, 256);
  const int selfBlks = cdiv_i((long long)N * 32, 256);
  const int eeBlks   = cdiv_i(Ee, 8);
  const int ekBlks   = cdiv_i(Ek, 8);
  const int gemmBlks = N / 16;                       // exact: N % 16 == 0

  // --- symmetric norms for both graphs ---
  k_deg_init <<<nodeBlks, 256, 0, stream>>>(dinv_e, N);
  k_deg_init <<<nodeBlks, 256, 0, stream>>>(dinv_k, N);
  k_deg_edges<<<cdiv_i(Ee, 256), 256, 0, stream>>>(e_s, e_r, dinv_e, Ee);
  k_deg_edges<<<cdiv_i(Ek, 256), 256, 0, stream>>>(k_s, k_r, dinv_k, Ek);
  k_deg_rsqrt<<<nodeBlks, 256, 0, stream>>>(dinv_e, N);
  k_deg_rsqrt<<<nodeBlks, 256, 0, stream>>>(dinv_k, N);

  // --- z branch: h = x@W_f2+b ; z0 = prop^2(h) ; zk = prop_knn(h) ; blend ---
  k_gemm_bias<<<gemmBlks, 256, 0, stream>>>(x, W_f2, b_f2, ping);
  k_prop_self <<<selfBlks, 256, 0, stream>>>(ping, dinv_e, pong, N);
  k_prop_edges<<<eeBlks,   256, 0, stream>>>(e_s, e_r, dinv_e, ping, pong, Ee);
  k_prop_self <<<selfBlks, 256, 0, stream>>>(pong, dinv_e, out_z0, N);
  k_prop_edges<<<eeBlks,   256, 0, stream>>>(e_s, e_r, dinv_e, pong, out_z0, Ee);
  k_prop_self <<<selfBlks, 256, 0, stream>>>(ping, dinv_k, pong, N);    // zk -> pong
  k_prop_edges<<<ekBlks,   256, 0, stream>>>(k_s, k_r, dinv_k, ping, pong, Ek);
  k_z1<<<cdiv_i(N, 8), 256, 0, stream>>>(out_z0, pong, W_lin, b_lin, out_z1, N);

  // --- h0 branch: h = x@W_s1+b ; h0 = prop^2(h) ---
  k_gemm_bias<<<gemmBlks, 256, 0, stream>>>(x, W_s1, b_s1, ping);
  k_prop_self <<<selfBlks, 256, 0, stream>>>(ping, dinv_e, pong, N);
  k_prop_edges<<<eeBlks,   256, 0, stream>>>(e_s, e_r, dinv_e, ping, pong, Ee);
  k_prop_self <<<selfBlks, 256, 0, stream>>>(pong, dinv_e, out_h0, N);
  k_prop_edges<<<eeBlks,   256, 0, stream>>>(e_s, e_r, dinv_e, pong, out_h0, Ee);

  // --- h1 branch: hop0 = x@W_l1+b ; 10 propagations ; sigmoid-scored mix ---
  k_gemm_bias<<<gemmBlks, 256, 0, stream>>>(x, W_l1, b_l1, pps);
  for (int k = 1; k <= KH; ++k) {
    const float* src = pps + (size_t)(k - 1) * NHf;
    float*       dst = pps + (size_t)k * NHf;
    k_prop_self <<<selfBlks, 256, 0, stream>>>(src, dinv_e, dst, N);
    k_prop_edges<<<eeBlks,   256, 0, stream>>>(e_s, e_r, dinv_e, src, dst, Ee);
  }
  k_score  <<<cdiv_i((long long)N * (KH + 1), 8), 256, 0, stream>>>(pps, W_p, b_p, score, N);
  k_combine<<<N, 128, 0, stream>>>(pps, score, out_h1, N);
}